// STMetaGCGRUEncoder_67920612819188
// MI455X (gfx1250) — compile-verified
//
#include <hip/hip_runtime.h>
#include <math.h>

// ---- problem constants (from reference) ----
#define Bb    16
#define Tt    12
#define Nn    325
#define Dd    1
#define Kk    3
#define Hh    64
#define MIi   127          // ST + Z = 95 + 32
#define LHh   128
#define BN    (Bb * Nn)    // 5200
#define COMBW (Dd + Hh)    // 65
#define SUPW  (Kk * COMBW) // 195
#define ODG   (SUPW * 2 * Hh)  // 24960
#define ODU   (SUPW * Hh)      // 12480

#define LDSTR 68           // LDS row stride (floats): 68%64=4 -> conflict-free,
                           // 68*4=272 bytes -> rows stay 16B aligned

typedef __attribute__((ext_vector_type(2))) float v2f;
typedef __attribute__((ext_vector_type(8))) float v8f;

// Low 32 bits of a flat shared pointer are the LDS byte offset (ISA 10.2:
// LDS aperture).
__device__ __forceinline__ unsigned lds_off(const void* p)
{
    return (unsigned)(unsigned long long)p;
}

// GLOBAL_LOAD_ASYNC_TO_LDS_B32 (GV mode), tracked by ASYNCcnt.
__device__ __forceinline__ void async_ld_b32(unsigned lds, const float* g)
{
    asm volatile("global_load_async_to_lds_b32 %0, %1, off"
                 :: "v"(lds), "v"(g) : "memory");
}

__device__ __forceinline__ void wait_async0()
{
    asm volatile("s_wait_asynccnt 0x0" ::: "memory");
}

// ---------------------------------------------------------------------------
// Generic tiled fp32 WMMA GEMM:  C = A(MxK) @ B(KxN) [+ bias[col]] [+ relu]
// 128 threads = 4 wave32 waves; block tile = 16 (M) x 64 (N); K staged in
// 64-wide LDS chunks, DOUBLE-BUFFERED via global_load_async_to_lds_b32 +
// s_wait_asynccnt; each wave runs 16x V_WMMA_F32_16X16X4_F32 per chunk.
// Ragged edge tiles fall back to guarded synchronous staging (zero fill).
// blockIdx.z decomposes as (zb, zk) with per-operand pointer strides so the
// same kernel serves the batched graph convolution (z = b*K + k).
// mode: 0 = none, 1 = +bias, 2 = +bias then relu
// ---------------------------------------------------------------------------
__global__ void __launch_bounds__(128)
wmma_gemm_f32(const float* __restrict__ A, int lda, long sAk, long sAb,
              const float* __restrict__ Bm, int ldb, long sBk, long sBb,
              float* __restrict__ C, int ldc, long sCk, long sCb,
              const float* __restrict__ bias,
              int M, int Ncols, int Kdim, int KB, int mode)
{
    __shared__ float sA[2][16][LDSTR];
    __shared__ float sB[2][64][LDSTR];

    const int zk = blockIdx.z % KB;
    const int zb = blockIdx.z / KB;
    A  += (long)zk * sAk + (long)zb * sAb;
    Bm += (long)zk * sBk + (long)zb * sBb;
    C  += (long)zk * sCk + (long)zb * sCb;

    const int tid  = threadIdx.x;
    const int wave = tid >> 5;
    const int lane = tid & 31;
    const int half = lane >> 4;
    const int lm   = lane & 15;

    const int m0 = blockIdx.y * 16;
    const int n0 = blockIdx.x * 64;

    // ---- tile staging (async fast path / guarded sync fallback) ----
    auto stageA = [&](int buf, int kc) {
        if ((m0 + 16 <= M) && (kc + 64 <= Kdim)) {
            #pragma unroll
            for (int jj = 0; jj < 8; ++jj) {
                int e = jj * 128 + tid;           // lane-consecutive columns
                int row = e >> 6, col = e & 63;
                async_ld_b32(lds_off(&sA[buf][row][col]),
                             A + (long)(m0 + row) * lda + (kc + col));
            }
        } else {
            int e0 = tid * 8;
            int row = e0 >> 6, col = e0 & 63;
            bool rowok = (m0 + row) < M;
            const float* src = A + (long)(m0 + row) * lda + (kc + col);
            #pragma unroll
            for (int j = 0; j < 8; ++j) {
                int k = kc + col + j;
                sA[buf][row][col + j] = (rowok && k < Kdim) ? src[j] : 0.0f;
            }
        }
    };
    auto stageB = [&](int buf, int kc) {
        if ((kc + 64 <= Kdim) && (n0 + 64 <= Ncols)) {
            #pragma unroll
            for (int jj = 0; jj < 32; ++jj) {
                int e = jj * 128 + tid;
                int row = e >> 6, col = e & 63;
                async_ld_b32(lds_off(&sB[buf][row][col]),
                             Bm + (long)(kc + row) * ldb + (n0 + col));
            }
        } else {
            #pragma unroll
            for (int blk = 0; blk < 4; ++blk) {
                int e = (blk * 128 + tid) * 8;
                int row = e >> 6, col = e & 63;
                int kr  = kc + row;
                const float* src = Bm + (long)kr * ldb + (n0 + col);
                bool kok = kr < Kdim;
                #pragma unroll
                for (int j = 0; j < 8; ++j) {
                    int n = n0 + col + j;
                    sB[buf][row][col + j] = (kok && n < Ncols) ? src[j] : 0.0f;
                }
            }
        }
    };

    v8f acc = {};
    const int nChunks = (Kdim + 63) >> 6;

    stageA(0, 0);
    stageB(0, 0);
    wait_async0();
    __syncthreads();

    int cur = 0;
    for (int c = 0; c < nChunks; ++c) {
        // prefetch next chunk into the other buffer while we compute
        if (c + 1 < nChunks) {
            stageA(cur ^ 1, (c + 1) * 64);
            stageB(cur ^ 1, (c + 1) * 64);
        }

        const float (*pA)[LDSTR] = sA[cur];
        const float (*pB)[LDSTR] = sB[cur];
        const int c0 = wave * 16;
        #pragma unroll
        for (int kk = 0; kk < 16; ++kk) {
            const int kb = kk * 4 + half * 2;
            v2f a, b;
            a.x = pA[lm][kb + 0];
            a.y = pA[lm][kb + 1];
            b.x = pB[kb + 0][c0 + lm];
            b.y = pB[kb + 1][c0 + lm];
            acc = __builtin_amdgcn_wmma_f32_16x16x4_f32(
                false, a, false, b, (short)0, acc, false, false);
        }

        wait_async0();      // prefetched tile has landed in LDS
        __syncthreads();    // + everyone done reading the old buffer
        cur ^= 1;
    }

    // D layout: lane half -> rows {v} / {8+v}, lm -> column
    const int colg = n0 + wave * 16 + lm;
    if (colg < Ncols) {
        const float bv = (mode >= 1) ? bias[colg] : 0.0f;
        #pragma unroll
        for (int v = 0; v < 8; ++v) {
            int row = m0 + half * 8 + v;
            if (row < M) {
                float val = acc[v] + bv;
                if (mode == 2) val = fmaxf(val, 0.0f);
                C[(long)row * ldc + colg] = val;
            }
        }
    }
}

// ---------------------------------------------------------------------------
// comb[b,n,:] = [ x[b,t,n,0], h[b,n,0..H-1] ]
// ---------------------------------------------------------------------------
__global__ void build_comb(const float* __restrict__ x, const float* __restrict__ h,
                           float* __restrict__ comb, int t)
{
    const int bn = blockIdx.x;
    const int b  = bn / Nn, n = bn % Nn;
    const int j  = threadIdx.x;
    if (j == 0)
        comb[(long)bn * COMBW] = x[(long)(b * Tt + t) * Nn + n];
    else if (j < COMBW)
        comb[(long)bn * COMBW + j] = h[(long)bn * Hh + (j - 1)];
}

// ---------------------------------------------------------------------------
// Gate phase: zr = sigmoid(sup(195) @ Wg[bn](195x128) + bg[bn]); z -> zbuf,
// cand = [x_t, r * h]. One block per (b,n); thread j owns output column j.
// Streaming Wg is the HBM-bound part -> coalesced row reads + prefetch.
// ---------------------------------------------------------------------------
__global__ void __launch_bounds__(128)
gate_kernel(const float* __restrict__ sup, const float* __restrict__ Wg,
            const float* __restrict__ bg, const float* __restrict__ x,
            const float* __restrict__ h, float* __restrict__ zbuf,
            float* __restrict__ cand, int t)
{
    __shared__ float s_sup[SUPW];
    __shared__ float s_r[Hh];
    const int bn = blockIdx.x;
    const int b  = bn / Nn, n = bn % Nn;
    const int j  = threadIdx.x;

    for (int i = j; i < SUPW; i += 128)
        s_sup[i] = sup[(long)bn * SUPW + i];
    __syncthreads();

    const float* W = Wg + (long)bn * SUPW * (2 * Hh);
    float acc = bg[(long)bn * (2 * Hh) + j];
    for (int i = 0; i < SUPW; ++i) {
        if ((i & 31) == 0)
            __builtin_prefetch(&W[(long)(i + 64) * (2 * Hh) + j], 0, 0);
        acc = fmaf(s_sup[i], W[(long)i * (2 * Hh) + j], acc);
    }
    float zr = 1.0f / (1.0f + __expf(-acc));
    if (j < Hh) zbuf[(long)bn * Hh + j] = zr;   // z
    else        s_r[j - Hh] = zr;               // r
    __syncthreads();

    if (j == 0)
        cand[(long)bn * COMBW] = x[(long)(b * Tt + t) * Nn + n];
    if (j < Hh)
        cand[(long)bn * COMBW + 1 + j] = s_r[j] * h[(long)bn * Hh + j];
}

// ---------------------------------------------------------------------------
// Update phase: nvec = tanh(supc @ Wu[bn] + bu[bn]); h = z*n + (1-z)*h;
// write hs[b,t,n,:] (and h_last on the final step).
// ---------------------------------------------------------------------------
__global__ void __launch_bounds__(64)
update_kernel(const float* __restrict__ sup, const float* __restrict__ Wu,
              const float* __restrict__ bu, const float* __restrict__ zbuf,
              float* __restrict__ h, float* __restrict__ hs,
              float* __restrict__ hlast, int t)
{
    __shared__ float s_sup[SUPW];
    const int bn = blockIdx.x;
    const int b  = bn / Nn, n = bn % Nn;
    const int j  = threadIdx.x;  // 0..63

    for (int i = j; i < SUPW; i += 64)
        s_sup[i] = sup[(long)bn * SUPW + i];
    __syncthreads();

    const float* W = Wu + (long)bn * SUPW * Hh;
    float acc = bu[(long)bn * Hh + j];
    for (int i = 0; i < SUPW; ++i) {
        if ((i & 31) == 0)
            __builtin_prefetch(&W[(long)(i + 64) * Hh + j], 0, 0);
        acc = fmaf(s_sup[i], W[(long)i * Hh + j], acc);
    }
    const float nv   = tanhf(acc);
    const float z    = zbuf[(long)bn * Hh + j];
    const float hold = h[(long)bn * Hh + j];
    const float hn   = z * nv + (1.0f - z) * hold;
    h[(long)bn * Hh + j] = hn;
    hs[((long)(b * Tt + t) * Nn + n) * Hh + j] = hn;
    if (t == Tt - 1) hlast[(long)bn * Hh + j] = hn;
}

__global__ void zero_kernel(float* __restrict__ p, long nelem)
{
    long i = (long)blockIdx.x * blockDim.x + threadIdx.x;
    if (i < nelem) p[i] = 0.0f;
}

// ---------------------------------------------------------------------------
extern "C" void kernel_launch(void* const* d_in, const int* in_sizes, int n_in,
                              void* d_out, int out_size, void* d_ws, size_t ws_size,
                              hipStream_t stream)
{
    (void)in_sizes; (void)n_in; (void)out_size; (void)ws_size;

    const float* G      = (const float*)d_in[0];
    const float* x      = (const float*)d_in[1];
    const float* meta   = (const float*)d_in[2];
    const float* lwg_w1 = (const float*)d_in[3];
    const float* lwg_b1 = (const float*)d_in[4];
    const float* lwg_w2 = (const float*)d_in[5];
    const float* lwg_b2 = (const float*)d_in[6];
    const float* lbg_w1 = (const float*)d_in[7];
    const float* lbg_b1 = (const float*)d_in[8];
    const float* lbg_w2 = (const float*)d_in[9];
    const float* lbg_b2 = (const float*)d_in[10];
    const float* lwu_w1 = (const float*)d_in[11];
    const float* lwu_b1 = (const float*)d_in[12];
    const float* lwu_w2 = (const float*)d_in[13];
    const float* lwu_b2 = (const float*)d_in[14];
    const float* lbu_w1 = (const float*)d_in[15];
    const float* lbu_b1 = (const float*)d_in[16];
    const float* lbu_w2 = (const float*)d_in[17];
    const float* lbu_b2 = (const float*)d_in[18];

    float* out   = (float*)d_out;
    float* hs    = out;                               // (B,T,N,H)
    float* hlast = out + (long)Bb * Tt * Nn * Hh;     // (B,N,H)

    // workspace carve-up (floats)
    float* ws = (float*)d_ws;
    long off = 0;
    float* Wg   = ws + off; off += (long)BN * ODG;    // 5200 x 24960 (519 MB)
    float* Wu   = ws + off; off += (long)BN * ODU;    // 5200 x 12480 (260 MB)
    float* bg   = ws + off; off += (long)BN * 2 * Hh;
    float* bu   = ws + off; off += (long)BN * Hh;
    float* hid  = ws + off; off += (long)BN * LHh;    // reused for all 4 MLPs
    float* h    = ws + off; off += (long)BN * Hh;
    float* comb = ws + off; off += (long)BN * COMBW;  // also reused for cand
    float* sup  = ws + off; off += (long)BN * SUPW;   // also reused for supc
    float* zb   = ws + off; off += (long)BN * Hh;

    const dim3 blk(128);
    auto gemm = [&](const float* A, int lda, long sAk, long sAb,
                    const float* Bm, int ldb, long sBk, long sBb,
                    float* C, int ldc, long sCk, long sCb,
                    const float* bias, int M, int Nc, int Kd, int KB,
                    int mode, int gz) {
        dim3 grid((Nc + 63) / 64, (M + 15) / 16, gz);
        hipLaunchKernelGGL(wmma_gemm_f32, grid, blk, 0, stream,
                           A, lda, sAk, sAb, Bm, ldb, sBk, sBb,
                           C, ldc, sCk, sCb, bias, M, Nc, Kd, KB, mode);
    };

    // h0 = 0
    {
        long n = (long)BN * Hh;
        hipLaunchKernelGGL(zero_kernel, dim3((unsigned)((n + 255) / 256)),
                           dim3(256), 0, stream, h, n);
    }

    // ---- meta-MLPs: hid = relu(meta@w1+b1); out = hid@w2+b2 ----
    gemm(meta, MIi, 0, 0, lwg_w1, LHh, 0, 0, hid, LHh, 0, 0, lwg_b1,
         BN, LHh, MIi, 1, 2, 1);
    gemm(hid, LHh, 0, 0, lwg_w2, ODG, 0, 0, Wg, ODG, 0, 0, lwg_b2,
         BN, ODG, LHh, 1, 1, 1);

    gemm(meta, MIi, 0, 0, lbg_w1, LHh, 0, 0, hid, LHh, 0, 0, lbg_b1,
         BN, LHh, MIi, 1, 2, 1);
    gemm(hid, LHh, 0, 0, lbg_w2, 2 * Hh, 0, 0, bg, 2 * Hh, 0, 0, lbg_b2,
         BN, 2 * Hh, LHh, 1, 1, 1);

    gemm(meta, MIi, 0, 0, lwu_w1, LHh, 0, 0, hid, LHh, 0, 0, lwu_b1,
         BN, LHh, MIi, 1, 2, 1);
    gemm(hid, LHh, 0, 0, lwu_w2, ODU, 0, 0, Wu, ODU, 0, 0, lwu_b2,
         BN, ODU, LHh, 1, 1, 1);

    gemm(meta, MIi, 0, 0, lbu_w1, LHh, 0, 0, hid, LHh, 0, 0, lbu_b1,
         BN, LHh, MIi, 1, 2, 1);
    gemm(hid, LHh, 0, 0, lbu_w2, Hh, 0, 0, bu, Hh, 0, 0, lbu_b2,
         BN, Hh, LHh, 1, 1, 1);

    // ---- GRU scan over T steps ----
    for (int t = 0; t < Tt; ++t) {
        hipLaunchKernelGGL(build_comb, dim3(BN), dim3(128), 0, stream,
                           x, h, comb, t);
        // sup[b,n,k,i] = sum_m G[k,n,m] * comb[b,m,i]   (z = b*K + k)
        gemm(G, Nn, (long)Nn * Nn, 0,
             comb, COMBW, 0, (long)Nn * COMBW,
             sup, SUPW, COMBW, (long)Nn * SUPW,
             nullptr, Nn, COMBW, Nn, Kk, 0, Bb * Kk);
        hipLaunchKernelGGL(gate_kernel, dim3(BN), dim3(128), 0, stream,
                           sup, Wg, bg, x, h, zb, comb /*cand*/, t);
        gemm(G, Nn, (long)Nn * Nn, 0,
             comb, COMBW, 0, (long)Nn * COMBW,
             sup, SUPW, COMBW, (long)Nn * SUPW,
             nullptr, Nn, COMBW, Nn, Kk, 0, Bb * Kk);
        hipLaunchKernelGGL(update_kernel, dim3(BN), dim3(64), 0, stream,
                           sup, Wu, bu, zb, h, hs, hlast, t);
    }
}